// FullAdultModel_26474178412845
// MI455X (gfx1250) — compile-verified
//
#include <hip/hip_runtime.h>

#define N_NEURONS 100000
#define N_EDGES   5000000
#define N_DM      512
#define BATCH     16

typedef __attribute__((ext_vector_type(2))) float v2f;
typedef __attribute__((ext_vector_type(8))) float v8f;

// h0[n*16+b] = x[n*16+b] * tw[ct[n]] + tb[ct[n]]
__global__ void init_h_kernel(const float* __restrict__ x,
                              const float* __restrict__ tw,
                              const float* __restrict__ tb,
                              const int* __restrict__ ct,
                              float* __restrict__ h, int total) {
    int t = blockIdx.x * blockDim.x + threadIdx.x;
    if (t >= total) return;
    int n = t >> 4;
    int c = ct[n];
    h[t] = fmaf(x[t], tw[c], tb[c]);
}

__global__ void zero_kernel(float* __restrict__ p, int total) {
    int t = blockIdx.x * blockDim.x + threadIdx.x;
    if (t < total) p[t] = 0.0f;
}

// One hop of h' = A @ h. 16 lanes per edge (lane = batch element):
// gather of src[col*16 + b] and atomic scatter to dst[row*16 + b] are each
// one coalesced 64B transaction per edge. Edge data + h buffers fit in the
// 192MB L2, so hops 2-4 are L2-resident.
__global__ void edge_hop_kernel(const float* __restrict__ ew,
                                const int* __restrict__ er,
                                const int* __restrict__ ec,
                                const float* __restrict__ src,
                                float* __restrict__ dst, int totalThreads) {
    int t = blockIdx.x * blockDim.x + threadIdx.x;
    if (t >= totalThreads) return;
    int e = t >> 4;
    int b = t & 15;
    float w = ew[e];
    int col = ec[e];
    int row = er[e];
    float v = w * src[col * BATCH + b];
    atomicAdd(&dst[row * BATCH + b], v);
}

// out[b] = sum_d (dm_vals[d]*fc_w[d]) * h[dm_cols[d], b] + fc_b
// Single wave32; 128 accumulating V_WMMA_F32_16X16X4_F32 ops (K=4 chunks).
// A row M=0 carries the weight vector (other rows zero via lane select,
// no branch -> EXEC stays all-ones as WMMA requires). B holds gathered h
// rows. D row 0 = VGPR0 on lanes 0-15 = the 16 batch outputs.
__global__ void final_fc_wmma_kernel(const float* __restrict__ h,
                                     const float* __restrict__ dm_vals,
                                     const int* __restrict__ dm_cols,
                                     const float* __restrict__ fc_w,
                                     const float* __restrict__ fc_b,
                                     float* __restrict__ out) {
    int lane = threadIdx.x;      // 0..31
    int n    = lane & 15;        // N (batch) index
    int half = lane >> 4;        // 0: K=0,1  1: K=2,3 (per ISA 16x4 f32 layout)
    v8f acc = {};
    for (int c = 0; c < N_DM / 4; ++c) {
        int kb = 4 * c + 2 * half;
        float g0 = dm_vals[kb]     * fc_w[kb];
        float g1 = dm_vals[kb + 1] * fc_w[kb + 1];
        int i0 = dm_cols[kb];
        int i1 = dm_cols[kb + 1];
        v2f a;
        a.x = (n == 0) ? g0 : 0.0f;   // only A row M=0 is non-zero
        a.y = (n == 0) ? g1 : 0.0f;
        v2f bb;
        bb.x = h[i0 * BATCH + n];     // B[k, n] = H[kb + k, n]
        bb.y = h[i1 * BATCH + n];
        acc = __builtin_amdgcn_wmma_f32_16x16x4_f32(
            /*neg_a=*/false, a, /*neg_b=*/false, bb,
            /*c_mod=*/(short)0, acc, /*reuse_a=*/false, /*reuse_b=*/false);
    }
    if (lane < 16) out[n] = acc[0] + fc_b[0];
}

extern "C" void kernel_launch(void* const* d_in, const int* in_sizes, int n_in,
                              void* d_out, int out_size, void* d_ws, size_t ws_size,
                              hipStream_t stream) {
    const float* x   = (const float*)d_in[0];   // [N_NEURONS, 16]
    const float* tw  = (const float*)d_in[1];   // [64]
    const float* tb  = (const float*)d_in[2];   // [64]
    const float* ew  = (const float*)d_in[3];   // [E]
    const float* dmv = (const float*)d_in[4];   // [512]
    const float* fcw = (const float*)d_in[5];   // [1,512]
    const float* fcb = (const float*)d_in[6];   // [1]
    const int* ct    = (const int*)d_in[7];     // [N_NEURONS]
    const int* er    = (const int*)d_in[8];     // [E]
    const int* ec    = (const int*)d_in[9];     // [E]
    const int* dmc   = (const int*)d_in[10];    // [512]
    float* out = (float*)d_out;                 // [16]

    float* hA = (float*)d_ws;
    float* hB = hA + (size_t)N_NEURONS * BATCH;

    const int totalH = N_NEURONS * BATCH;   // 1.6M
    const int totalE = N_EDGES * BATCH;     // 80M (fits int)
    const int blk = 256;

    init_h_kernel<<<(totalH + blk - 1) / blk, blk, 0, stream>>>(x, tw, tb, ct, hA, totalH);

    float* src = hA;
    float* dst = hB;
    for (int l = 0; l < 4; ++l) {
        zero_kernel<<<(totalH + blk - 1) / blk, blk, 0, stream>>>(dst, totalH);
        edge_hop_kernel<<<(totalE + blk - 1) / blk, blk, 0, stream>>>(ew, er, ec, src, dst, totalE);
        float* tmp = src; src = dst; dst = tmp;
    }

    final_fc_wmma_kernel<<<1, 32, 0, stream>>>(src, dmv, dmc, fcw, fcb, out);
}